// HyperPINN_50981261804289
// MI455X (gfx1250) — compile-verified
//
#include <hip/hip_runtime.h>
#include <math.h>

typedef __attribute__((ext_vector_type(2))) float v2f;
typedef __attribute__((ext_vector_type(8))) float v8f;
typedef __attribute__((ext_vector_type(4))) int v4i;

#define NL 64
#define NSF 64
#define GDIM 96
#define NPTS (GDIM * GDIM * GDIM) /* 884736 */
#define WS_STRIDE 640             /* per-batch param floats: 384 w0 | 64 b0 | 70 w1 | 1 b1 */
#define OMEGA 30.0f
#define WAVES 8
#define GROUPS 8 /* 16-point WMMA groups per wave: 8*8*16 = 1024 pts/block, 884736/1024 = 864 blocks */

/* lane-xor shuffle via ds_swizzle (offset[14:10]=xor, [4:0]=and) */
#define SWZ_XOR(v, m) \
    __int_as_float(__builtin_amdgcn_ds_swizzle(__float_as_int(v), (((m) << 10) | 0x1f)))

#if defined(__AMDGCN__) && __has_builtin(__builtin_amdgcn_global_load_async_to_lds_b128)
#define HAS_ASYNC 1
/* signature (from hipcc diagnostic): (int4*, int4*, imm offset, cpol) */
#define ASYNC_CP128(gsrc, ldst) \
    __builtin_amdgcn_global_load_async_to_lds_b128((v4i*)(gsrc), (v4i*)(ldst), 0, 0)
#else
#define HAS_ASYNC 0
#endif

#if defined(__AMDGCN__) && __has_builtin(__builtin_amdgcn_s_wait_asynccnt)
#define WAIT_ASYNC(n) __builtin_amdgcn_s_wait_asynccnt(n)
#else
#define WAIT_ASYNC(n) asm volatile("s_wait_asynccnt " #n ::: "memory")
#endif

// ---------------- per-batch hypernetwork param precompute ----------------
__global__ void hyper_params_kernel(const float* __restrict__ h,
                                    const float* __restrict__ W0w, const float* __restrict__ b0w,
                                    const float* __restrict__ W0b, const float* __restrict__ b0b,
                                    const float* __restrict__ W1w, const float* __restrict__ b1w,
                                    const float* __restrict__ W1b, const float* __restrict__ b1b,
                                    float* __restrict__ ws) {
    __shared__ float hs[NL];
    const int b = blockIdx.x;
    const int t = threadIdx.x;
    if (t < NL) hs[t] = h[b * NL + t];
    __syncthreads();
    for (int j = t; j < 519; j += blockDim.x) {
        const float* Wcol;
        float bias;
        int ncol, col;
        if (j < 384)      { col = j;       Wcol = W0w; bias = b0w[col]; ncol = 384; }
        else if (j < 448) { col = j - 384; Wcol = W0b; bias = b0b[col]; ncol = 64;  }
        else if (j < 518) { col = j - 448; Wcol = W1w; bias = b1w[col]; ncol = 70;  }
        else              { col = 0;       Wcol = W1b; bias = b1b[0];   ncol = 1;   }
        float acc = bias;
#pragma unroll 8
        for (int k = 0; k < NL; ++k) acc += hs[k] * Wcol[k * ncol + col];
        ws[b * WS_STRIDE + j] = acc;
    }
}

// ---------------- main: async-staged feats -> WMMA layer0 -> sin -> layer1 dot ----------------
__global__ __launch_bounds__(256, 2) void hyper_main_kernel(const float* __restrict__ x,
                                                            const float* __restrict__ ws,
                                                            float* __restrict__ out) {
    const int batch = blockIdx.y;
    const int lane = threadIdx.x & 31;
    const int wave = threadIdx.x >> 5;
    const int half = lane >> 4; /* A/B matrices interleave K over lane halves */
    const int lm16 = lane & 15;

    const float* P = ws + batch * WS_STRIDE;

    // ---- hoisted per-lane params ----
    // B-matrix tiles (w0 is (6,64) row-major). 16x16x4 f32 B layout:
    // VGPR0 = K0 (lanes 0-15) / K2 (lanes 16-31); VGPR1 = K1 / K3.
    v2f Blo[4], Bhi[4];
    float biasv[4], w1v[4];
#pragma unroll
    for (int t2 = 0; t2 < 4; ++t2) {
        const int n = 16 * t2 + lm16;
        Blo[t2][0] = P[(half ? 2 : 0) * NSF + n];
        Blo[t2][1] = P[(half ? 3 : 1) * NSF + n];
        Bhi[t2][0] = half ? 0.0f : P[4 * NSF + n]; /* K=6 pad */
        Bhi[t2][1] = half ? 0.0f : P[5 * NSF + n]; /* K=7 pad */
        biasv[t2] = P[384 + n];
        w1v[t2] = P[448 + 6 + n];
    }
    float w1f[6];
#pragma unroll
    for (int k = 0; k < 6; ++k) w1f[k] = P[448 + k];
    const float b1 = P[518];

    __shared__ float outbuf[WAVES * 16];
#if HAS_ASYNC
    __shared__ float xbuf[2][WAVES][48]; /* double-buffered 16-pt coord tiles, 192 B/wave */
#endif

    const long long batchBase = (long long)batch * NPTS;
    const int wgroup0 = (blockIdx.x * WAVES + wave) * GROUPS;
    const float* gx0 = x + (batchBase + (long long)wgroup0 * 16) * 3;

#if HAS_ASYNC
    /* prologue: stage group 0 (12 lanes x b128 = 192 B), ASYNCcnt++ */
    if (lane < 12) ASYNC_CP128(gx0 + lane * 4, &xbuf[0][wave][lane * 4]);
#endif

#pragma unroll 1
    for (int g = 0; g < GROUPS; ++g) {
        const int pbase = (wgroup0 + g) * 16;

#if HAS_ASYNC
        if (g + 1 < GROUPS) {
            if (lane < 12)
                ASYNC_CP128(gx0 + (g + 1) * 48 + lane * 4, &xbuf[(g + 1) & 1][wave][lane * 4]);
            WAIT_ASYNC(1); /* in-order completion: oldest (this group's tile) is done */
        } else {
            WAIT_ASYNC(0);
        }
        asm volatile("" ::: "memory");
        const float* lp = &xbuf[g & 1][wave][lm16 * 3];
        const float xc = lp[0] * OMEGA;
        const float yc = lp[1] * OMEGA;
        const float zc = lp[2] * OMEGA;
#else
        const float* xp = x + (batchBase + pbase + lm16) * 3;
        __builtin_prefetch(xp + 16 * 3, 0, 0); /* next group -> global_prefetch */
        const float xc = xp[0] * OMEGA;
        const float yc = xp[1] * OMEGA;
        const float zc = xp[2] * OMEGA;
#endif
        const float r = __builtin_amdgcn_sqrtf(xc * xc + yc * yc);
        const float inv = __builtin_amdgcn_rcpf(r + 1.0f); /* v_rcp_f32, not IEEE div chain */
        const float sn = xc * inv;
        const float cs = yc * inv;

        // A-matrix 16x4 f32 layout: VGPR0 = K0|K2, VGPR1 = K1|K3 across lane halves
        v2f Alo, Ahi;
        Alo[0] = half ? zc : xc;   /* K0 / K2 */
        Alo[1] = half ? r : yc;    /* K1 / K3 */
        Ahi[0] = half ? 0.0f : sn; /* K4 / K6(pad) */
        Ahi[1] = half ? 0.0f : cs; /* K5 / K7(pad) */

        float partial[8];
#pragma unroll
        for (int rr = 0; rr < 8; ++rr) partial[rr] = 0.0f;

#pragma unroll
        for (int t2 = 0; t2 < 4; ++t2) {
            v8f c;
#pragma unroll
            for (int rr = 0; rr < 8; ++rr) c[rr] = biasv[t2]; /* bias folded into C */
            c = __builtin_amdgcn_wmma_f32_16x16x4_f32(false, Alo, false, Blo[t2],
                                                      (short)0, c, false, false);
            c = __builtin_amdgcn_wmma_f32_16x16x4_f32(false, Ahi, false, Bhi[t2],
                                                      (short)0, c, false, false);
#pragma unroll
            for (int rr = 0; rr < 8; ++rr) {
                const float s = __builtin_amdgcn_sinf(c[rr]); /* v_sin_f32 == sin(2*pi*x) */
                partial[rr] += s * w1v[t2];
            }
        }

        // feats . w1[0:6] + b1 for point lane%16
        const float fc = b1 + xc * w1f[0] + yc * w1f[1] + zc * w1f[2] + r * w1f[3] +
                         sn * w1f[4] + cs * w1f[5];

#pragma unroll
        for (int rr = 0; rr < 8; ++rr) {
            float v = partial[rr];
            v += SWZ_XOR(v, 1); /* reduce over N within each 16-lane half */
            v += SWZ_XOR(v, 2);
            v += SWZ_XOR(v, 4);
            v += SWZ_XOR(v, 8);
            const int mRow = rr + (half << 3); /* D layout: VGPR rr = row rr / rr+8 */
            v += __int_as_float(__builtin_amdgcn_ds_bpermute(mRow << 2, __float_as_int(fc)));
            if (lm16 == 0) outbuf[wave * 16 + mRow] = v;
        }
        asm volatile("s_wait_dscnt 0" ::: "memory"); /* intra-wave LDS RAW */
        if (lane < 16) out[batchBase + pbase + lane] = outbuf[wave * 16 + lane];
    }
}

extern "C" void kernel_launch(void* const* d_in, const int* in_sizes, int n_in,
                              void* d_out, int out_size, void* d_ws, size_t ws_size,
                              hipStream_t stream) {
    const float* h = (const float*)d_in[0];
    const float* x = (const float*)d_in[1];
    const float* W0w = (const float*)d_in[2];
    const float* b0w = (const float*)d_in[3];
    const float* W0b = (const float*)d_in[4];
    const float* b0b = (const float*)d_in[5];
    const float* W1w = (const float*)d_in[6];
    const float* b1w = (const float*)d_in[7];
    const float* W1b = (const float*)d_in[8];
    const float* b1b = (const float*)d_in[9];
    float* out = (float*)d_out;
    float* ws = (float*)d_ws;

    hipLaunchKernelGGL(hyper_params_kernel, dim3(2), dim3(256), 0, stream,
                       h, W0w, b0w, W0b, b0b, W1w, b1w, W1b, b1b, ws);

    dim3 grid(NPTS / (WAVES * GROUPS * 16), 2, 1); /* 864 x 2, exact */
    hipLaunchKernelGGL(hyper_main_kernel, grid, dim3(256), 0, stream, x, ws, out);
}